// MTSPNet_9655086481585
// MI455X (gfx1250) — compile-verified
//
#include <hip/hip_runtime.h>
#include <math.h>

typedef __attribute__((ext_vector_type(2))) float v2f;
typedef __attribute__((ext_vector_type(8))) float v8f;

#define BB 4
#define MM 8
#define NN 128
#define DD 64
#define SA_EPS 1e-8f

// ---------------------------------------------------------------------------
// Kernel 1: three dense f32 GEMMs via V_WMMA_F32_16X16X4_F32.
// One wave (32 lanes) per 16x16 output tile; K=64 in 16 steps of K=4.
//   tiles 0..127   : hf = cities[b] @ Wf   (128x64 @ 64x64), 4b * 8rt * 4ct
//   tiles 128..255 : ht = cities[b] @ Wt
//   tiles 256..271 : hg = groups[b] @ Wg   (8x64 @ 64x64, M padded to 16)
// A rows beyond the real matrix are CLAMPED (duplicate last row) instead of
// masked, so the K-loop has no EXEC divergence; the store mask discards the
// padded rows. EXEC is all-1s at every WMMA as the ISA requires.
// ---------------------------------------------------------------------------
__global__ __launch_bounds__(32)
void mtsp_gemm_wmma(const float* __restrict__ cities,
                    const float* __restrict__ groups,
                    const float* __restrict__ W1,
                    float* __restrict__ hg,
                    float* __restrict__ hf,
                    float* __restrict__ ht)
{
    int ti   = blockIdx.x;
    int lane = threadIdx.x;

    const float* A; const float* W; float* Out; int arows, rt, ct, b;
    if (ti < 128) {                      // hf
        b = ti >> 5; int r = ti & 31; rt = r >> 2; ct = r & 3;
        A = cities + b * NN * DD; arows = NN;
        W = W1 + 64 * DD;                // Wf rows 64..127 of W1 (192x64 row-major)
        Out = hf + b * NN * DD;
    } else if (ti < 256) {               // ht
        int t = ti - 128;
        b = t >> 5; int r = t & 31; rt = r >> 2; ct = r & 3;
        A = cities + b * NN * DD; arows = NN;
        W = W1 + 128 * DD;               // Wt rows 128..191
        Out = ht + b * NN * DD;
    } else {                             // hg (M=8 padded to 16)
        int t = ti - 256;
        b = t >> 2; ct = t & 3; rt = 0;
        A = groups + b * MM * DD; arows = MM;
        W = W1;                          // Wg rows 0..63
        Out = hg + b * MM * DD;
    }

    int half = lane >> 4;                // 0: K pair {k,k+1}; 1: {k+2,k+3}
    int l16  = lane & 15;
    int arow = rt * 16 + l16;            // logical A row this lane carries
    int bcol = ct * 16 + l16;            // B column this lane carries

    // Clamp instead of mask: no EXEC divergence inside the K-loop.
    const float* arowp = A + min(arow, arows - 1) * DD;

    v8f acc = {};
    #pragma unroll
    for (int kk = 0; kk < DD; kk += 4) {
        int ka = kk + half * 2;
        v2f a  = *(const v2f*)(arowp + ka);                        // A[row, ka..ka+1]
        v2f bv;
        bv.x = W[(ka + 0) * DD + bcol];                            // B[ka,   col]
        bv.y = W[(ka + 1) * DD + bcol];                            // B[ka+1, col]
        acc = __builtin_amdgcn_wmma_f32_16x16x4_f32(
                  false, a, false, bv, (short)0, acc, false, false);
    }
    // D layout: VGPR i -> rows i (lanes 0-15) and i+8 (lanes 16-31), col = lane%16
    #pragma unroll
    for (int i = 0; i < 8; ++i) {
        int row = rt * 16 + i + half * 8;
        if (row < arows) Out[row * DD + bcol] = acc[i];
    }
}

// ---------------------------------------------------------------------------
// Kernel 2: fused MLP. Never materializes the (4,8,128,128,64) tensor.
// out[b,m,f,t] = sum_e relu(hg[b,m,e]+b1[e] + hf[b,f,e] + ht[b,t,e]) * W2[e] + b2
// One block per (b, m, 16-row f-tile); ht/hf tiles staged in LDS with +1 pad
// to avoid 64-bank conflicts on the stride-64 rows.
// ---------------------------------------------------------------------------
__global__ __launch_bounds__(256)
void mtsp_mlp_out(const float* __restrict__ hg, const float* __restrict__ hf,
                  const float* __restrict__ ht, const float* __restrict__ b1,
                  const float* __restrict__ W2, const float* __restrict__ b2,
                  float* __restrict__ out)
{
    __shared__ float cvec[DD];
    __shared__ float w2s[DD];
    __shared__ float hfs[16 * 65];
    __shared__ float hts[NN * 65];

    int tid = threadIdx.x;
    int b = blockIdx.z, m = blockIdx.y, f0 = blockIdx.x * 16;

    if (tid < DD) {
        cvec[tid] = hg[(b * MM + m) * DD + tid] + b1[tid];
        w2s[tid]  = W2[tid];
    }
    for (int idx = tid; idx < 16 * DD; idx += 256) {
        int f = idx >> 6, e = idx & 63;
        hfs[f * 65 + e] = hf[(b * NN + f0 + f) * DD + e];
    }
    for (int idx = tid; idx < NN * DD; idx += 256) {
        int t = idx >> 6, e = idx & 63;
        hts[t * 65 + e] = ht[(b * NN + t) * DD + e];
    }
    __syncthreads();

    float bias2 = b2[0];
    #pragma unroll
    for (int k = 0; k < 8; ++k) {           // 16*128 outputs / 256 threads
        int idx = tid + k * 256;
        int f = idx >> 7, t = idx & 127;
        const float* hfr = &hfs[f * 65];
        const float* htr = &hts[t * 65];
        float acc = 0.f;
        #pragma unroll 8
        for (int e = 0; e < DD; ++e)
            acc += fmaxf(cvec[e] + hfr[e] + htr[e], 0.f) * w2s[e];
        out[((b * MM + m) * NN + f0 + f) * NN + t] = acc + bias2;
    }
}

// ---------------------------------------------------------------------------
// Kernel 3: softassign prep: subtract per-structure max, exp, in place.
// One workgroup per batch b; all reductions internal to the workgroup.
//   depots_max[m]  = max_j x[b,m,0,j]
//   others_max[i]  = max_{m,j} x[b,m,i,j]   (i >= 1)
// ---------------------------------------------------------------------------
__global__ __launch_bounds__(256)
void mtsp_sa_prep(float* __restrict__ x)
{
    __shared__ float partial[MM * NN];   // per-(m,i) row max
    __shared__ float others[NN];

    int tid = threadIdx.x, b = blockIdx.x;
    float* xb = x + b * MM * NN * NN;

    for (int r = tid; r < MM * NN; r += 256) {      // r = m*128 + i
        int m = r >> 7, i = r & 127;
        const float4* row = (const float4*)(xb + (m * NN + i) * NN);
        float mx = -INFINITY;
        for (int j = 0; j < NN / 4; ++j) {
            float4 v = row[j];
            mx = fmaxf(mx, fmaxf(fmaxf(v.x, v.y), fmaxf(v.z, v.w)));
        }
        partial[r] = mx;
    }
    __syncthreads();
    if (tid >= 1 && tid < NN) {
        float mx = partial[tid];
        for (int m = 1; m < MM; ++m) mx = fmaxf(mx, partial[m * NN + tid]);
        others[tid] = mx;
    }
    __syncthreads();
    for (int r = tid; r < MM * NN; r += 256) {
        int m = r >> 7, i = r & 127;
        float mx = (i == 0) ? partial[m * NN] : others[i];
        float4* row = (float4*)(xb + (m * NN + i) * NN);
        for (int j = 0; j < NN / 4; ++j) {
            float4 v = row[j];
            v.x = expf(v.x - mx); v.y = expf(v.y - mx);
            v.z = expf(v.z - mx); v.w = expf(v.w - mx);
            row[j] = v;
        }
    }
}

// ---------------------------------------------------------------------------
// Kernel 4: one softassign layer. Clamp to EPS, normalize row 0 per (b,m)
// over columns, rows 1..127 per (b,row) over (m, columns), then write the
// result TRANSPOSED into y: y[b,m,j,i] = max(x[b,m,i,j],EPS)/denom.
// One workgroup per batch; 2 MB total stays L2-resident.
// ---------------------------------------------------------------------------
__global__ __launch_bounds__(256)
void mtsp_sa_iter(const float* __restrict__ x, float* __restrict__ y)
{
    __shared__ float partial[MM * NN];   // per-(m,i) row sum of clamped values
    __shared__ float nsum[NN];           // per-i sum over m (i >= 1)

    int tid = threadIdx.x, b = blockIdx.x;
    const float* xb = x + b * MM * NN * NN;
    float* yb = y + b * MM * NN * NN;

    for (int r = tid; r < MM * NN; r += 256) {
        int m = r >> 7, i = r & 127;
        const float4* row = (const float4*)(xb + (m * NN + i) * NN);
        float s = 0.f;
        for (int j = 0; j < NN / 4; ++j) {
            float4 v = row[j];
            s += fmaxf(v.x, SA_EPS) + fmaxf(v.y, SA_EPS)
               + fmaxf(v.z, SA_EPS) + fmaxf(v.w, SA_EPS);
        }
        partial[r] = s;
    }
    __syncthreads();
    if (tid >= 1 && tid < NN) {
        float s = 0.f;
        for (int m = 0; m < MM; ++m) s += partial[m * NN + tid];
        nsum[tid] = s;
    }
    __syncthreads();
    for (int r = tid; r < MM * NN; r += 256) {
        int m = r >> 7, i = r & 127;
        float inv = 1.0f / ((i == 0) ? partial[m * NN] : nsum[i]);
        const float* row = xb + (m * NN + i) * NN;
        float* col = yb + m * NN * NN + i;          // y[m][j][i]
        for (int j = 0; j < NN; ++j)
            col[j * NN] = fmaxf(row[j], SA_EPS) * inv;
    }
}

// ---------------------------------------------------------------------------
extern "C" void kernel_launch(void* const* d_in, const int* in_sizes, int n_in,
                              void* d_out, int out_size, void* d_ws, size_t ws_size,
                              hipStream_t stream)
{
    const float* cities = (const float*)d_in[0];
    const float* groups = (const float*)d_in[1];
    const float* W1     = (const float*)d_in[2];
    const float* b1     = (const float*)d_in[3];
    const float* W2     = (const float*)d_in[4];
    const float* b2     = (const float*)d_in[5];

    float* ws = (float*)d_ws;
    float* hg = ws;                        //  4*8*64    = 2048
    float* hf = hg + BB * MM * DD;         //  4*128*64  = 32768
    float* ht = hf + BB * NN * DD;         //  32768
    float* P0 = ht + BB * NN * DD;         //  4*8*128*128 = 524288
    float* P1 = P0 + BB * MM * NN * NN;    //  524288   (total ~4.3 MB)

    mtsp_gemm_wmma<<<272, 32, 0, stream>>>(cities, groups, W1, hg, hf, ht);
    mtsp_mlp_out<<<dim3(NN / 16, MM, BB), 256, 0, stream>>>(hg, hf, ht, b1, W2, b2, P0);
    mtsp_sa_prep<<<BB, 256, 0, stream>>>(P0);

    float* bufs[2] = { P0, P1 };
    for (int it = 0; it < 10; ++it) {
        const float* src = bufs[it & 1];
        float* dst = (it == 9) ? (float*)d_out : bufs[(it + 1) & 1];
        mtsp_sa_iter<<<BB, 256, 0, stream>>>(src, dst);
    }
}